// GNNFraudDetector_58016418234844
// MI455X (gfx1250) — compile-verified
//
#include <hip/hip_runtime.h>
#include <math.h>

typedef __attribute__((ext_vector_type(2))) float v2f;
typedef __attribute__((ext_vector_type(8))) float v8f;

#define GNN_N 100000
#define GNN_E 1600000
#define GNN_D 64

// ---------------- utility: zero fill ----------------
__global__ void k_fill_zero(float* __restrict__ p, long n) {
    long i = (long)blockIdx.x * blockDim.x + threadIdx.x;
    long stride = (long)gridDim.x * blockDim.x;
    for (; i < n; i += stride) p[i] = 0.0f;
}

// ---------------- degree count ----------------
__global__ void k_count(const int* __restrict__ dst, float* __restrict__ cnt, int ne) {
    int i = blockIdx.x * blockDim.x + threadIdx.x;
    if (i < ne) unsafeAtomicAdd(&cnt[dst[i]], 1.0f);
}

__global__ void k_inv(float* __restrict__ cnt, int n) {
    int i = blockIdx.x * blockDim.x + threadIdx.x;
    if (i < n) {
        float c = cnt[i];
        cnt[i] = 1.0f / fmaxf(c, 1.0f);
    }
}

// ---------------- edge scatter: agg[dst] += h[src] ----------------
// one wave per edge; lane handles 2 contiguous floats (coalesced 256B row)
__global__ void k_scatter(const float* __restrict__ h, const int* __restrict__ src,
                          const int* __restrict__ dst, float* __restrict__ agg, int ne) {
    long t = (long)blockIdx.x * blockDim.x + threadIdx.x;
    long e = t >> 5;
    if (e >= ne) return;
    int lane = (int)(t & 31);
    int s = src[e];   // wave-uniform (scalarized)
    int d = dst[e];
    float2 v = *(const float2*)(h + (long)s * GNN_D + lane * 2);
    float* ap = agg + (long)d * GNN_D + lane * 2;
    unsafeAtomicAdd(ap, v.x);
    unsafeAtomicAdd(ap + 1, v.y);
}

// ---------------- fused SAGE layer GEMM ----------------
// hout = relu( (agg .* inv_row) @ Wl + bl + hin @ Wr )
// one wave per 16-row tile; 4 x 16-col output tiles held in v8f accumulators;
// K=64 walked in steps of 4 with V_WMMA_F32_16X16X4_F32 (fp32 in/out).
__global__ void __launch_bounds__(256)
k_sage(const float* __restrict__ agg, const float* __restrict__ inv,
       const float* __restrict__ hin, const float* __restrict__ Wl,
       const float* __restrict__ bl, const float* __restrict__ Wr,
       float* __restrict__ hout, int n_tiles) {
    int wave = threadIdx.x >> 5;
    int lane = threadIdx.x & 31;
    int tile = blockIdx.x * 8 + wave;
    if (tile >= n_tiles) return;          // wave-uniform: EXEC stays all-1s

    int row0 = tile * 16;
    int rc   = lane & 15;                 // A row index AND B column index for this lane
    int koff = (lane >> 4) * 2;           // K sub-offset: lanes 0-15 -> K,K+1 ; 16-31 -> K+2,K+3

    float scale = inv[row0 + rc];         // mean = agg / max(cnt,1)
    const float* aggRow = agg + (long)(row0 + rc) * GNN_D;
    const float* hinRow = hin + (long)(row0 + rc) * GNN_D;

    v8f acc[4];
    acc[0] = (v8f){}; acc[1] = (v8f){}; acc[2] = (v8f){}; acc[3] = (v8f){};

    #pragma unroll
    for (int kk = 0; kk < 16; ++kk) {
        int kA = kk * 4 + koff;
        // A tiles (16x4 fp32): a.x = A[row][kA], a.y = A[row][kA+1]
        v2f a1 = *(const v2f*)(aggRow + kA);
        a1.x *= scale; a1.y *= scale;
        v2f a2 = *(const v2f*)(hinRow + kA);

        #pragma unroll
        for (int t = 0; t < 4; ++t) {
            int cb = t * 16 + rc;
            // B tiles (4x16 fp32): b.x = W[kA][col], b.y = W[kA+1][col]
            v2f bL, bR;
            bL.x = Wl[(long)kA * GNN_D + cb];
            bL.y = Wl[(long)(kA + 1) * GNN_D + cb];
            bR.x = Wr[(long)kA * GNN_D + cb];
            bR.y = Wr[(long)(kA + 1) * GNN_D + cb];
            acc[t] = __builtin_amdgcn_wmma_f32_16x16x4_f32(
                false, a1, false, bL, (short)0, acc[t], false, false);
            acc[t] = __builtin_amdgcn_wmma_f32_16x16x4_f32(
                false, a2, false, bR, (short)0, acc[t], false, false);
        }
    }

    // Epilogue: bias + relu + store.
    // C/D layout: VGPR r, lanes 0-15 -> row r, lanes 16-31 -> row r+8, col = lane&15
    int rhalf = (lane >> 4) * 8;
    #pragma unroll
    for (int t = 0; t < 4; ++t) {
        int c = t * 16 + rc;
        float bb = bl[c];
        #pragma unroll
        for (int rr = 0; rr < 8; ++rr) {
            float v = acc[t][rr] + bb;
            v = v > 0.0f ? v : 0.0f;
            hout[(long)(row0 + rr + rhalf) * GNN_D + c] = v;
        }
    }
}

// ---------------- MLP head: sigmoid(relu(h@Wf1+bf1)@Wf2+bf2) ----------------
__global__ void k_head(const float* __restrict__ h, const float* __restrict__ Wf1,
                       const float* __restrict__ bf1, const float* __restrict__ Wf2,
                       const float* __restrict__ bf2, float* __restrict__ out, int n) {
    int i = blockIdx.x * blockDim.x + threadIdx.x;
    if (i >= n) return;
    float hid[32];
    #pragma unroll
    for (int j = 0; j < 32; ++j) hid[j] = bf1[j];
    const float* hr = h + (long)i * GNN_D;
    #pragma unroll 4
    for (int k = 0; k < GNN_D; ++k) {
        float hk = hr[k];
        const float* w = Wf1 + k * 32;   // uniform address -> scalar loads
        #pragma unroll
        for (int j = 0; j < 32; ++j) hid[j] = fmaf(hk, w[j], hid[j]);
    }
    float z = bf2[0];
    #pragma unroll
    for (int j = 0; j < 32; ++j) z += fmaxf(hid[j], 0.0f) * Wf2[j];
    out[i] = 1.0f / (1.0f + expf(-z));
}

extern "C" void kernel_launch(void* const* d_in, const int* in_sizes, int n_in,
                              void* d_out, int out_size, void* d_ws, size_t ws_size,
                              hipStream_t stream) {
    const float* x   = (const float*)d_in[0];
    const int*   ei  = (const int*)  d_in[1];
    const float* W1l = (const float*)d_in[2];
    const float* b1l = (const float*)d_in[3];
    const float* W1r = (const float*)d_in[4];
    const float* W2l = (const float*)d_in[5];
    const float* b2l = (const float*)d_in[6];
    const float* W2r = (const float*)d_in[7];
    const float* Wf1 = (const float*)d_in[8];
    const float* bf1 = (const float*)d_in[9];
    const float* Wf2 = (const float*)d_in[10];
    const float* bf2 = (const float*)d_in[11];
    float* out = (float*)d_out;

    const int N = GNN_N;
    const int E = GNN_E;
    const int* src = ei;
    const int* dst = ei + E;

    // workspace layout (aligned, ~94 MB total)
    char* ws = (char*)d_ws;
    float* inv  = (float*)(ws);                               // N floats
    float* aggb = (float*)(ws + (1l  << 20));                 // N*64 floats (reused)
    float* h1   = (float*)(ws + (1l  << 20) + (32l << 20));   // N*64 floats
    float* h2   = (float*)(ws + (1l  << 20) + (64l << 20));   // N*64 floats

    const int n_tiles = N / 16;                // 6250, exact
    const int sage_blocks = (n_tiles + 7) / 8; // 8 waves per block
    const long scat_threads = (long)E * 32;
    const int scat_blocks = (int)((scat_threads + 255) / 256);

    // inverse degree (shared by both layers)
    k_fill_zero<<<1024, 256, 0, stream>>>(inv, N);
    k_count<<<(E + 255) / 256, 256, 0, stream>>>(dst, inv, E);
    k_inv<<<(N + 255) / 256, 256, 0, stream>>>(inv, N);

    // ---- layer 1 ----
    k_fill_zero<<<2048, 256, 0, stream>>>(aggb, (long)N * GNN_D);
    k_scatter<<<scat_blocks, 256, 0, stream>>>(x, src, dst, aggb, E);
    k_sage<<<sage_blocks, 256, 0, stream>>>(aggb, inv, x, W1l, b1l, W1r, h1, n_tiles);

    // ---- layer 2 ----
    k_fill_zero<<<2048, 256, 0, stream>>>(aggb, (long)N * GNN_D);
    k_scatter<<<scat_blocks, 256, 0, stream>>>(h1, src, dst, aggb, E);
    k_sage<<<sage_blocks, 256, 0, stream>>>(aggb, inv, h1, W2l, b2l, W2r, h2, n_tiles);

    // ---- head ----
    k_head<<<(N + 255) / 256, 256, 0, stream>>>(h2, Wf1, bf1, Wf2, bf2, out, N);
}